// D3Contrast_88716844466339
// MI455X (gfx1250) — compile-verified
//
#include <hip/hip_runtime.h>
#include <math.h>

// D3Contrast forward for MI455X (gfx1250, wave32).
// All dense matmuls use V_WMMA_F32_16X16X4_F32 (full fp32 precision; the
// problem is latency/scan bound, not compute bound, so no low-precision).
// GEMM: one wave computes a 16x64 tile (A fragment reused across 4 WMMAs).
// Bounds handling is branch-free: B/C column indices are clamped (an OOB
// column only pollutes its own, never-stored, output column); only the
// K-remainder step zero-masks the A fragment.

namespace {

constexpr int kB  = 4;
constexpr int kL  = 256;
constexpr int kM  = 55;
constexpr int kDM = 256;
constexpr int kCO = 55;
constexpr int kNS = 256;   // mamba state size N
constexpr float kPI = 3.14159265358979323846f;

typedef float v2f __attribute__((ext_vector_type(2)));
typedef float v8f __attribute__((ext_vector_type(8)));

// ---------------- reductions (wave32) ----------------
__device__ __forceinline__ float wave_sum(float v) {
#pragma unroll
  for (int off = 16; off > 0; off >>= 1) v += __shfl_down(v, off, 32);
  return v;
}
__device__ __forceinline__ float wave_max(float v) {
#pragma unroll
  for (int off = 16; off > 0; off >>= 1) v = fmaxf(v, __shfl_down(v, off, 32));
  return v;
}
__device__ __forceinline__ float block_sum(float v, float* sh) {
  const int lane = threadIdx.x & 31, w = threadIdx.x >> 5;
  v = wave_sum(v);
  if (lane == 0) sh[w] = v;
  __syncthreads();
  if (threadIdx.x == 0) {
    const int nw = (blockDim.x + 31) >> 5;
    float s = sh[0];
    for (int i = 1; i < nw; ++i) s += sh[i];
    sh[0] = s;
  }
  __syncthreads();
  float r = sh[0];
  __syncthreads();
  return r;
}
__device__ __forceinline__ float block_max(float v, float* sh) {
  const int lane = threadIdx.x & 31, w = threadIdx.x >> 5;
  v = wave_max(v);
  if (lane == 0) sh[w] = v;
  __syncthreads();
  if (threadIdx.x == 0) {
    const int nw = (blockDim.x + 31) >> 5;
    float s = sh[0];
    for (int i = 1; i < nw; ++i) s = fmaxf(s, sh[i]);
    sh[0] = s;
  }
  __syncthreads();
  float r = sh[0];
  __syncthreads();
  return r;
}
__device__ __forceinline__ float siluf(float x) { return x / (1.f + expf(-x)); }

// ---------------- generic WMMA f32 GEMM ----------------
// C[b][row][col] = alpha * sum_k A[b][row][k] * B(k,col)  (+ bias[col]) (+ add)
// transB=1 : B(k,col) = W[col*ldw + k]   (A @ W^T)
// transB=0 : B(k,col) = W[k*ldw + col]   (A @ W, used for attn @ v)
// REQUIREMENT: rows % 16 == 0 (true for every call here: 1024 or 256).
// One wave (32 threads) per block; tile = 16 rows x 64 cols.
__global__ void wmma_gemm_kernel(const float* __restrict__ A, const float* __restrict__ W,
                                 const float* __restrict__ bias, const float* __restrict__ addm,
                                 float* __restrict__ C,
                                 int rows, int K, int N,
                                 int lda, int ldw, int ldc,
                                 long sA, long sW, long sC,
                                 int transB, float alpha) {
  const int bidx = blockIdx.z;
  const float* Ab = A + (long)bidx * sA;
  const float* Wb = W + (long)bidx * sW;
  float*       Cb = C + (long)bidx * sC;

  const int lane = (int)(threadIdx.x & 31u);
  const int half = lane >> 4;   // selects K pair {0,1} vs {2,3}
  const int r    = lane & 15;   // A: row within tile, B: col within tile
  const int row0 = blockIdx.x * 16;
  const int col0 = blockIdx.y * 64;

  const int arow = row0 + r;                 // always in range (rows%16==0)
  const float* __restrict__ Arow = Ab + (long)arow * lda;

  int  bcol[4];
  int  bcolc[4];                             // clamped: OOB cols never stored
#pragma unroll
  for (int j = 0; j < 4; ++j) {
    bcol[j]  = col0 + 16 * j + r;
    bcolc[j] = bcol[j] < N ? bcol[j] : (N - 1);
  }

  v8f acc[4];
#pragma unroll
  for (int j = 0; j < 4; ++j) acc[j] = (v8f){0.f,0.f,0.f,0.f,0.f,0.f,0.f,0.f};

  const int Kmain = K & ~3;
  if (transB) {
    const float* __restrict__ Bb0 = Wb + (long)bcolc[0] * ldw;
    const float* __restrict__ Bb1 = Wb + (long)bcolc[1] * ldw;
    const float* __restrict__ Bb2 = Wb + (long)bcolc[2] * ldw;
    const float* __restrict__ Bb3 = Wb + (long)bcolc[3] * ldw;
    for (int k0 = 0; k0 < Kmain; k0 += 4) {
      const int ka = k0 + 2 * half;
      v2f a, b0, b1, b2, b3;
      a.x  = Arow[ka];  a.y  = Arow[ka + 1];
      b0.x = Bb0[ka];   b0.y = Bb0[ka + 1];
      b1.x = Bb1[ka];   b1.y = Bb1[ka + 1];
      b2.x = Bb2[ka];   b2.y = Bb2[ka + 1];
      b3.x = Bb3[ka];   b3.y = Bb3[ka + 1];
      acc[0] = __builtin_amdgcn_wmma_f32_16x16x4_f32(false, a, false, b0, (short)0, acc[0], false, false);
      acc[1] = __builtin_amdgcn_wmma_f32_16x16x4_f32(false, a, false, b1, (short)0, acc[1], false, false);
      acc[2] = __builtin_amdgcn_wmma_f32_16x16x4_f32(false, a, false, b2, (short)0, acc[2], false, false);
      acc[3] = __builtin_amdgcn_wmma_f32_16x16x4_f32(false, a, false, b3, (short)0, acc[3], false, false);
    }
    if (Kmain < K) {  // remainder: zero-mask A only (0 * garbage == 0)
      const int ka = Kmain + 2 * half;
      const bool ok0 = ka < K, ok1 = (ka + 1) < K;
      const int ka0 = ok0 ? ka : (K - 1);
      const int ka1 = ok1 ? (ka + 1) : (K - 1);
      v2f a, b0, b1, b2, b3;
      float t0 = Arow[ka0], t1 = Arow[ka1];
      a.x = ok0 ? t0 : 0.f;  a.y = ok1 ? t1 : 0.f;
      b0.x = Bb0[ka0];  b0.y = Bb0[ka1];
      b1.x = Bb1[ka0];  b1.y = Bb1[ka1];
      b2.x = Bb2[ka0];  b2.y = Bb2[ka1];
      b3.x = Bb3[ka0];  b3.y = Bb3[ka1];
      acc[0] = __builtin_amdgcn_wmma_f32_16x16x4_f32(false, a, false, b0, (short)0, acc[0], false, false);
      acc[1] = __builtin_amdgcn_wmma_f32_16x16x4_f32(false, a, false, b1, (short)0, acc[1], false, false);
      acc[2] = __builtin_amdgcn_wmma_f32_16x16x4_f32(false, a, false, b2, (short)0, acc[2], false, false);
      acc[3] = __builtin_amdgcn_wmma_f32_16x16x4_f32(false, a, false, b3, (short)0, acc[3], false, false);
    }
  } else {
    const float* __restrict__ Bc0 = Wb + bcolc[0];
    const float* __restrict__ Bc1 = Wb + bcolc[1];
    const float* __restrict__ Bc2 = Wb + bcolc[2];
    const float* __restrict__ Bc3 = Wb + bcolc[3];
    for (int k0 = 0; k0 < Kmain; k0 += 4) {
      const int ka = k0 + 2 * half;
      const long o0 = (long)ka * ldw, o1 = (long)(ka + 1) * ldw;
      v2f a, b0, b1, b2, b3;
      a.x  = Arow[ka];  a.y  = Arow[ka + 1];
      b0.x = Bc0[o0];   b0.y = Bc0[o1];
      b1.x = Bc1[o0];   b1.y = Bc1[o1];
      b2.x = Bc2[o0];   b2.y = Bc2[o1];
      b3.x = Bc3[o0];   b3.y = Bc3[o1];
      acc[0] = __builtin_amdgcn_wmma_f32_16x16x4_f32(false, a, false, b0, (short)0, acc[0], false, false);
      acc[1] = __builtin_amdgcn_wmma_f32_16x16x4_f32(false, a, false, b1, (short)0, acc[1], false, false);
      acc[2] = __builtin_amdgcn_wmma_f32_16x16x4_f32(false, a, false, b2, (short)0, acc[2], false, false);
      acc[3] = __builtin_amdgcn_wmma_f32_16x16x4_f32(false, a, false, b3, (short)0, acc[3], false, false);
    }
    if (Kmain < K) {
      const int ka = Kmain + 2 * half;
      const bool ok0 = ka < K, ok1 = (ka + 1) < K;
      const long o0 = (long)(ok0 ? ka : (K - 1)) * ldw;
      const long o1 = (long)(ok1 ? (ka + 1) : (K - 1)) * ldw;
      const int ka0 = ok0 ? ka : (K - 1);
      const int ka1 = ok1 ? (ka + 1) : (K - 1);
      v2f a, b0, b1, b2, b3;
      float t0 = Arow[ka0], t1 = Arow[ka1];
      a.x = ok0 ? t0 : 0.f;  a.y = ok1 ? t1 : 0.f;
      b0.x = Bc0[o0];  b0.y = Bc0[o1];
      b1.x = Bc1[o0];  b1.y = Bc1[o1];
      b2.x = Bc2[o0];  b2.y = Bc2[o1];
      b3.x = Bc3[o0];  b3.y = Bc3[o1];
      acc[0] = __builtin_amdgcn_wmma_f32_16x16x4_f32(false, a, false, b0, (short)0, acc[0], false, false);
      acc[1] = __builtin_amdgcn_wmma_f32_16x16x4_f32(false, a, false, b1, (short)0, acc[1], false, false);
      acc[2] = __builtin_amdgcn_wmma_f32_16x16x4_f32(false, a, false, b2, (short)0, acc[2], false, false);
      acc[3] = __builtin_amdgcn_wmma_f32_16x16x4_f32(false, a, false, b3, (short)0, acc[3], false, false);
    }
  }

  // D layout: lane r = col within tile, VGPR v -> row = v + 8*half
#pragma unroll
  for (int j = 0; j < 4; ++j) {
    if (bcol[j] < N) {
      const float bv = bias ? bias[bcol[j]] : 0.f;
#pragma unroll
      for (int v = 0; v < 8; ++v) {
        const int row = row0 + v + 8 * half;
        float val = alpha * acc[j][v] + bv;
        if (addm) val += addm[(long)bidx * sC + (long)row * ldc + bcol[j]];
        Cb[(long)row * ldc + bcol[j]] = val;
      }
    }
  }
}

// ---------------- constants on device ----------------
__global__ void init_pe_kernel(float* pe, int Lseq, int D) {
  int idx = blockIdx.x * blockDim.x + threadIdx.x;
  if (idx >= Lseq * D) return;
  int l = idx / D, d = idx % D;
  int i2 = (d >> 1) * 2;  // even index used for div term
  float div = expf((float)i2 * (-logf(10000.f) / (float)D));
  float ang = (float)l * div;
  pe[idx] = (d & 1) ? cosf(ang) : sinf(ang);
}
__global__ void init_dct_kernel(float* C, float* CI, int D) {
  int idx = blockIdx.x * blockDim.x + threadIdx.x;
  if (idx >= D * D) return;
  int a = idx / D, b = idx % D;
  // C[k=a][n=b] = 2*cos(pi*(2n+1)*k/(2D))
  C[idx] = 2.f * cosf(kPI * (float)(2 * b + 1) * (float)a / (2.f * (float)D));
  // CI[n=a][k=b] = cos(pi*(2n+1)*k/(2D))/(2D), column k=0 halved
  float ci = cosf(kPI * (float)(2 * a + 1) * (float)b / (2.f * (float)D)) / (2.f * (float)D);
  if (b == 0) ci *= 0.5f;
  CI[idx] = ci;
}

// ---------------- pre-processing ----------------
// instance norm over L per (b,m)
__global__ void norm_time_kernel(const float* __restrict__ x, float* __restrict__ xn,
                                 int Lseq, int M) {
  __shared__ float sh[32];
  const int bm = blockIdx.x;
  const int b = bm / M, m = bm % M;
  const int l = threadIdx.x;
  const long idx = ((long)b * Lseq + l) * M + m;
  const float v = x[idx];
  const float mu = block_sum(v, sh) / (float)Lseq;
  const float df = v - mu;
  const float var = block_sum(df * df, sh) / (float)Lseq;
  xn[idx] = df * rsqrtf(var + 1e-5f);
}
// moving average (window 25, edge-replicated) -> trend_in, seasonal
__global__ void trend_split_kernel(const float* __restrict__ xn,
                                   float* __restrict__ trend_in,
                                   float* __restrict__ seasonal, int Lseq, int M) {
  __shared__ float row[256];
  const int bm = blockIdx.x;
  const int b = bm / M, m = bm % M;
  const int l = threadIdx.x;
  row[l] = xn[((long)b * Lseq + l) * M + m];
  __syncthreads();
  float s = 0.f;
#pragma unroll
  for (int j = -12; j <= 12; ++j) {
    int jj = l + j;
    jj = jj < 0 ? 0 : (jj >= Lseq ? Lseq - 1 : jj);
    s += row[jj];
  }
  const float t = s / 25.f;
  const long idx = ((long)b * Lseq + l) * M + m;
  trend_in[idx] = t;
  seasonal[idx] = row[l] - t;
}
// token conv (circular k=3 over L) + positional encoding
__global__ void token_conv_pe_kernel(const float* __restrict__ x, const float* __restrict__ w,
                                     const float* __restrict__ pe, float* __restrict__ out,
                                     int Lseq, int M, int D) {
  __shared__ float rows[3 * 64];
  const long bl = blockIdx.x;
  const int b = (int)(bl / Lseq), l = (int)(bl % Lseq);
  const int d = threadIdx.x;
  for (int idx = threadIdx.x; idx < 3 * M; idx += blockDim.x) {
    const int k = idx / M, m = idx % M;
    int li = l + k - 1;
    li = (li + Lseq) % Lseq;  // circular pad [x[-1], x, x[0]]
    rows[k * 64 + m] = x[((long)b * Lseq + li) * M + m];
  }
  __syncthreads();
  float s = pe[(long)l * D + d];
  for (int k = 0; k < 3; ++k)
    for (int m = 0; m < M; ++m) s += rows[k * 64 + m] * w[((long)d * M + m) * 3 + k];
  out[bl * D + d] = s;
}
__global__ void reverse_l_kernel(const float* __restrict__ in, float* __restrict__ out,
                                 int Lseq, int D) {
  const long bl = blockIdx.x;
  const int b = (int)(bl / Lseq), l = (int)(bl % Lseq);
  out[bl * D + threadIdx.x] = in[((long)b * Lseq + (Lseq - 1 - l)) * D + threadIdx.x];
}
__global__ void add2_kernel(const float* __restrict__ a, const float* __restrict__ b,
                            float* __restrict__ c, long n) {
  long i = (long)blockIdx.x * blockDim.x + threadIdx.x;
  if (i < n) c[i] = a[i] + b[i];
}

// ---------------- mamba pieces ----------------
// depthwise causal conv (k=4) + SiLU : u[b,l,e]
__global__ void dconv_silu_kernel(const float* __restrict__ xz, const float* __restrict__ cw,
                                  const float* __restrict__ cb, float* __restrict__ u,
                                  int ED, int ldxz, int Lseq) {
  const int e = blockIdx.x * blockDim.x + threadIdx.x;
  if (e >= ED) return;
  const long bl = blockIdx.y;
  const int b = (int)(bl / Lseq), l = (int)(bl % Lseq);
  float s = cb[e];
#pragma unroll
  for (int k = 0; k < 4; ++k) {
    const int li = l + k - 3;
    if (li >= 0) s += xz[((long)b * Lseq + li) * ldxz + e] * cw[e * 4 + k];
  }
  u[bl * ED + e] = siluf(s);
}
// delta = softplus(dbc[:, :rk] @ dt_w.T + dt_b)   (K = 4 or 16: plain kernel)
__global__ void delta_softplus_kernel(const float* __restrict__ dbc,
                                      const float* __restrict__ dtw,
                                      const float* __restrict__ dtb,
                                      float* __restrict__ delta, int ED, int rk, int ldd) {
  const int e = blockIdx.x * blockDim.x + threadIdx.x;
  if (e >= ED) return;
  const long row = blockIdx.y;
  const float* dr = dbc + row * ldd;
  float s = dtb[e];
  for (int r = 0; r < rk; ++r) s += dr[r] * dtw[e * rk + r];
  delta[row * ED + e] = (s > 20.f) ? s : log1pf(expf(s));
}
// selective scan: one block per (e, b); thread n holds h[b,e,n] in a register.
// y[b,t,e] = sum_n h * C[b,t,n] + u[b,t,e]*Dp[e]
__global__ void mamba_scan_kernel(const float* __restrict__ delta, const float* __restrict__ u,
                                  const float* __restrict__ dbc, const float* __restrict__ Alog,
                                  const float* __restrict__ Dp, float* __restrict__ y,
                                  int ED, int ldd, int bOff, int cOff, int Lseq) {
  __shared__ float red[8];
  const int e = blockIdx.x;
  const int b = blockIdx.y;
  const int n = threadIdx.x;                 // 0..255 (N state)
  const int lane = threadIdx.x & 31, w = threadIdx.x >> 5;
  const float a = -expf(Alog[(long)e * kNS + n]);  // A = -exp(Alog)
  const float dpe = Dp[e];
  float h = 0.f;
  for (int t = 0; t < Lseq; ++t) {
    const long row = (long)b * Lseq + t;
    const float d  = delta[row * ED + e];
    const float ut = u[row * ED + e];
    const float Bt = dbc[row * ldd + bOff + n];
    const float Ct = dbc[row * ldd + cOff + n];
    h = expf(d * a) * h + (d * ut) * Bt;
    float p = h * Ct;
#pragma unroll
    for (int off = 16; off > 0; off >>= 1) p += __shfl_down(p, off, 32);
    if (lane == 0) red[w] = p;
    __syncthreads();
    if (threadIdx.x == 0) {
      float s = red[0];
#pragma unroll
      for (int i = 1; i < 8; ++i) s += red[i];
      y[row * ED + e] = s + ut * dpe;
    }
    __syncthreads();
  }
}
// g = y * silu(z), z = xz[:, ED + e]
__global__ void gate_silu_kernel(const float* __restrict__ y, const float* __restrict__ xz,
                                 float* __restrict__ g, int ED, int ldxz) {
  const int e = blockIdx.x * blockDim.x + threadIdx.x;
  if (e >= ED) return;
  const long row = blockIdx.y;
  const float z = xz[row * ldxz + ED + e];
  g[row * ED + e] = y[row * ED + e] * siluf(z);
}

// ---------------- heads ----------------
__global__ void gelu_ln_kernel(const float* __restrict__ xin, const float* __restrict__ g,
                               const float* __restrict__ bb, float* __restrict__ out, int D) {
  __shared__ float sh[32];
  const long row = blockIdx.x;
  const int d = threadIdx.x;
  const float x = xin[row * D + d];
  const float ge = 0.5f * x * (1.f + erff(x * 0.70710678118654752f));  // exact gelu
  const float mu = block_sum(ge, sh) / (float)D;
  const float df = ge - mu;
  const float var = block_sum(df * df, sh) / (float)D;
  out[row * D + d] = df * rsqrtf(var + 1e-5f) * g[d] + bb[d];
}
__global__ void softmax_row_kernel(float* __restrict__ s, int D) {
  __shared__ float sh[32];
  const long row = blockIdx.x;
  const int j = threadIdx.x;
  const float v = s[row * D + j];
  const float m = block_max(v, sh);
  const float e = expf(v - m);
  const float sum = block_sum(e, sh);
  s[row * D + j] = e / sum;
}

}  // namespace

extern "C" void kernel_launch(void* const* d_in, const int* in_sizes, int n_in,
                              void* d_out, int out_size, void* d_ws, size_t ws_size,
                              hipStream_t stream) {
  (void)in_sizes; (void)n_in; (void)out_size; (void)ws_size;

  const float* x        = (const float*)d_in[0];
  const float* tok1_w   = (const float*)d_in[1];
  const float* tok2_w   = (const float*)d_in[2];
  const float* mt_in_w  = (const float*)d_in[3];
  const float* mt_cw    = (const float*)d_in[4];
  const float* mt_cb    = (const float*)d_in[5];
  const float* mt_xp_w  = (const float*)d_in[6];
  const float* mt_dt_w  = (const float*)d_in[7];
  const float* mt_dt_b  = (const float*)d_in[8];
  const float* mt_Alog  = (const float*)d_in[9];
  const float* mt_Dp    = (const float*)d_in[10];
  const float* mt_out_w = (const float*)d_in[11];
  const float* me_in_w  = (const float*)d_in[12];
  const float* me_cw    = (const float*)d_in[13];
  const float* me_cb    = (const float*)d_in[14];
  const float* me_xp_w  = (const float*)d_in[15];
  const float* me_dt_w  = (const float*)d_in[16];
  const float* me_dt_b  = (const float*)d_in[17];
  const float* me_Alog  = (const float*)d_in[18];
  const float* me_Dp    = (const float*)d_in[19];
  const float* me_out_w = (const float*)d_in[20];
  const float* ln_g     = (const float*)d_in[21];
  const float* ln_b     = (const float*)d_in[22];
  const float* wq = (const float*)d_in[23]; const float* bq = (const float*)d_in[24];
  const float* wk = (const float*)d_in[25]; const float* bk = (const float*)d_in[26];
  const float* wv = (const float*)d_in[27]; const float* bv = (const float*)d_in[28];
  const float* wo = (const float*)d_in[29]; const float* bo = (const float*)d_in[30];
  const float* p1_w = (const float*)d_in[31]; const float* p1_b = (const float*)d_in[32];
  const float* p2_w = (const float*)d_in[33]; const float* p2_b = (const float*)d_in[34];

  float* out_time = (float*)d_out;
  float* out_fre  = out_time + (long)kB * kL * kCO;

  const int rows = kB * kL;  // 1024, batch folded into rows for plain GEMMs

  // bump allocator on workspace (~36 MB total)
  float* wp = (float*)d_ws;
  auto alloc = [&](size_t n) { float* p = wp; wp += n; return p; };

  float* pe      = alloc((size_t)kL * kDM);
  float* cdct    = alloc((size_t)kDM * kDM);
  float* cidct   = alloc((size_t)kDM * kDM);
  float* xn      = alloc((size_t)rows * kM);
  float* trendin = alloc((size_t)rows * kM);
  float* season  = alloc((size_t)rows * kM);
  float* trend   = alloc((size_t)rows * kM);
  float* xz_mt   = alloc((size_t)rows * 220);
  float* u_mt    = alloc((size_t)rows * 110);
  float* dbc_mt  = alloc((size_t)rows * 516);
  float* dlt_mt  = alloc((size_t)rows * 110);
  float* y_mt    = alloc((size_t)rows * 110);
  float* g_mt    = alloc((size_t)rows * 110);
  float* x_new   = alloc((size_t)rows * kDM);
  float* xe      = alloc((size_t)rows * kDM);
  float* xe_rev  = alloc((size_t)rows * kDM);
  float* xz_me   = alloc((size_t)rows * 1024);
  float* u_me    = alloc((size_t)rows * 512);
  float* dbc_me  = alloc((size_t)rows * 528);
  float* dlt_me  = alloc((size_t)rows * 512);
  float* y_me    = alloc((size_t)rows * 512);
  float* g_me    = alloc((size_t)rows * 512);
  float* te1     = alloc((size_t)rows * kDM);
  float* te2     = alloc((size_t)rows * kDM);
  float* ln_out  = alloc((size_t)rows * kDM);
  float* freD    = alloc((size_t)rows * kDM);
  float* qb      = alloc((size_t)rows * kDM);
  float* kb      = alloc((size_t)rows * kDM);
  float* vb      = alloc((size_t)rows * kDM);
  float* attn    = alloc((size_t)rows * kDM);   // 4 x 256 x 256
  float* ao      = alloc((size_t)rows * kDM);
  float* t1      = alloc((size_t)rows * kDM);
  float* t2      = alloc((size_t)rows * kDM);

  auto gemm = [&](const float* A, int lda, const float* W, int ldw,
                  const float* bias, const float* addm, float* C, int ldc,
                  int rws, int K, int N, int transB, float alpha,
                  int batch, long sA, long sW, long sC) {
    dim3 grid((rws + 15) / 16, (N + 63) / 64, batch);
    wmma_gemm_kernel<<<grid, 32, 0, stream>>>(A, W, bias, addm, C, rws, K, N,
                                              lda, ldw, ldc, sA, sW, sC, transB, alpha);
  };

  // full mamba block (host sequencing)
  auto run_mamba = [&](const float* xin, int dm, int ED, int rk,
                       const float* in_w, const float* cw, const float* cb,
                       const float* xp_w, const float* dt_w, const float* dt_b,
                       const float* Alog, const float* Dp, const float* out_w,
                       float* xz, float* u, float* dbc, float* dlt, float* y, float* g,
                       float* out) {
    const int twoED = 2 * ED;
    const int nd = rk + 2 * kNS;
    gemm(xin, dm, in_w, dm, nullptr, nullptr, xz, twoED, rows, dm, twoED, 1, 1.f, 1, 0, 0, 0);
    dconv_silu_kernel<<<dim3((ED + 255) / 256, rows), 256, 0, stream>>>(xz, cw, cb, u, ED, twoED, kL);
    gemm(u, ED, xp_w, ED, nullptr, nullptr, dbc, nd, rows, ED, nd, 1, 1.f, 1, 0, 0, 0);
    delta_softplus_kernel<<<dim3((ED + 255) / 256, rows), 256, 0, stream>>>(dbc, dt_w, dt_b, dlt, ED, rk, nd);
    mamba_scan_kernel<<<dim3(ED, kB), kNS, 0, stream>>>(dlt, u, dbc, Alog, Dp, y, ED, nd, rk, rk + kNS, kL);
    gate_silu_kernel<<<dim3((ED + 255) / 256, rows), 256, 0, stream>>>(y, xz, g, ED, twoED);
    gemm(g, ED, out_w, ED, nullptr, nullptr, out, dm, rows, ED, dm, 1, 1.f, 1, 0, 0, 0);
  };

  // ---- constants ----
  init_pe_kernel<<<(kL * kDM + 255) / 256, 256, 0, stream>>>(pe, kL, kDM);
  init_dct_kernel<<<(kDM * kDM + 255) / 256, 256, 0, stream>>>(cdct, cidct, kDM);

  // ---- normalize + trend/seasonal split ----
  norm_time_kernel<<<kB * kM, kL, 0, stream>>>(x, xn, kL, kM);
  trend_split_kernel<<<kB * kM, kL, 0, stream>>>(xn, trendin, season, kL, kM);

  // ---- trend mamba (dm=55, ED=110, rk=4) ----
  run_mamba(trendin, kM, 110, 4, mt_in_w, mt_cw, mt_cb, mt_xp_w, mt_dt_w, mt_dt_b,
            mt_Alog, mt_Dp, mt_out_w, xz_mt, u_mt, dbc_mt, dlt_mt, y_mt, g_mt, trend);

  // ---- token convs + PE ----
  token_conv_pe_kernel<<<rows, kDM, 0, stream>>>(season, tok1_w, pe, x_new, kL, kM, kDM);
  token_conv_pe_kernel<<<rows, kDM, 0, stream>>>(xn, tok2_w, pe, xe, kL, kM, kDM);
  reverse_l_kernel<<<rows, kDM, 0, stream>>>(xe, xe_rev, kL, kDM);

  // ---- embed mamba x2 (dm=256, ED=512, rk=16) ----
  run_mamba(xe, kDM, 512, 16, me_in_w, me_cw, me_cb, me_xp_w, me_dt_w, me_dt_b,
            me_Alog, me_Dp, me_out_w, xz_me, u_me, dbc_me, dlt_me, y_me, g_me, te1);
  run_mamba(xe_rev, kDM, 512, 16, me_in_w, me_cw, me_cb, me_xp_w, me_dt_w, me_dt_b,
            me_Alog, me_Dp, me_out_w, xz_me, u_me, dbc_me, dlt_me, y_me, g_me, te2);
  {
    long n = (long)rows * kDM;
    add2_kernel<<<(int)((n + 255) / 256), 256, 0, stream>>>(te1, te2, te1, n);
  }

  // ---- time head: LN(gelu(time)) @ p1.T + p1_b ----
  gelu_ln_kernel<<<rows, kDM, 0, stream>>>(te1, ln_g, ln_b, ln_out, kDM);
  gemm(ln_out, kDM, p1_w, kDM, p1_b, nullptr, out_time, kCO, rows, kDM, kCO, 1, 1.f, 1, 0, 0, 0);

  // ---- frequency path: DCT -> attention -> IDCT -> proj (+trend) ----
  gemm(x_new, kDM, cdct, kDM, nullptr, nullptr, freD, kDM, rows, kDM, kDM, 1, 1.f, 1, 0, 0, 0);
  gemm(freD, kDM, wq, kDM, bq, nullptr, qb, kDM, rows, kDM, kDM, 1, 1.f, 1, 0, 0, 0);
  gemm(freD, kDM, wk, kDM, bk, nullptr, kb, kDM, rows, kDM, kDM, 1, 1.f, 1, 0, 0, 0);
  gemm(freD, kDM, wv, kDM, bv, nullptr, vb, kDM, rows, kDM, kDM, 1, 1.f, 1, 0, 0, 0);
  // scores[b] = q[b] @ k[b]^T / 16  (batched)
  {
    const long st = (long)kL * kDM;
    gemm(qb, kDM, kb, kDM, nullptr, nullptr, attn, kL, kL, kDM, kL, 1, 0.0625f, kB, st, st, st);
  }
  softmax_row_kernel<<<rows, kDM, 0, stream>>>(attn, kDM);
  // ao[b] = attn[b] @ v[b]  (batched, transB = 0)
  {
    const long st = (long)kL * kDM;
    gemm(attn, kL, vb, kDM, nullptr, nullptr, ao, kDM, kL, kL, kDM, 0, 1.f, kB, st, st, st);
  }
  gemm(ao, kDM, wo, kDM, bo, nullptr, t1, kDM, rows, kDM, kDM, 1, 1.f, 1, 0, 0, 0);
  gemm(t1, kDM, cidct, kDM, nullptr, nullptr, t2, kDM, rows, kDM, kDM, 1, 1.f, 1, 0, 0, 0);
  // fre = t2 @ p2.T + p2_b + trend
  gemm(t2, kDM, p2_w, kDM, p2_b, trend, out_fre, kCO, rows, kDM, kCO, 1, 1.f, 1, 0, 0, 0);
}